// GCN_LSTM_Model_52604759441752
// MI455X (gfx1250) — compile-verified
//
#include <hip/hip_runtime.h>
#include <math.h>

// ---------------- problem constants (match reference) ----------------
#define Bb   32
#define Tt   16
#define BT   (Bb * Tt)        // 512
#define Nn   1000
#define Cc   16
#define Gg   32
#define NG   (Nn * Gg)        // 32000
#define Hh   512
#define H4   (4 * Hh)         // 2048
#define Oo   8

typedef __attribute__((ext_vector_type(2))) float v2f;
typedef __attribute__((ext_vector_type(8))) float v8f;
typedef __attribute__((ext_vector_type(4))) int   v4i;

#if __has_builtin(__builtin_amdgcn_global_load_async_to_lds_b128)
#define HAVE_ASYNC_LDS 1
typedef __attribute__((address_space(1))) v4i* as1_v4i;   // global 16B vector ptr
typedef __attribute__((address_space(3))) v4i* as3_v4i;   // LDS 16B vector ptr
#else
#define HAVE_ASYNC_LDS 0
#endif

static __device__ __forceinline__ v8f wmma_f32(v2f a, v2f b, v8f c) {
    // D = A(16x4 f32) * B(4x16 f32) + C(16x16 f32)
    return __builtin_amdgcn_wmma_f32_16x16x4_f32(false, a, false, b, (short)0, c,
                                                 false, false);
}

// 16-byte global -> LDS copy; async (no VGPR round-trip, ASYNCcnt) when available
static __device__ __forceinline__ void cp16_to_lds(float* l, const float* g) {
#if HAVE_ASYNC_LDS
    __builtin_amdgcn_global_load_async_to_lds_b128((as1_v4i)(g), (as3_v4i)(l), 0, 0);
#else
    *(float4*)l = *(const float4*)g;
#endif
}

template <int N>
static __device__ __forceinline__ void wait_async_le() {
#if HAVE_ASYNC_LDS
#if __has_builtin(__builtin_amdgcn_s_wait_asynccnt)
    __builtin_amdgcn_s_wait_asynccnt(N);
#else
    asm volatile("s_wait_asynccnt %0" :: "i"(N) : "memory");
#endif
#endif
}

static __device__ __forceinline__ float sigm(float x) {
    return 1.0f / (1.0f + __expf(-x));
}

// ---------------- small utility kernels ----------------
__global__ void k_zero(float* __restrict__ p, int n) {
    int i = blockIdx.x * blockDim.x + threadIdx.x;
    if (i < n) p[i] = 0.0f;
}

__global__ void k_deg(const int* __restrict__ dst, const float* __restrict__ ew,
                      float* __restrict__ deg, int E) {
    int e = blockIdx.x * blockDim.x + threadIdx.x;
    if (e < E) atomicAdd(&deg[dst[e]], ew[e]);
}

__global__ void k_norm(const int* __restrict__ src, const int* __restrict__ dst,
                       const float* __restrict__ ew, const float* __restrict__ deg,
                       float* __restrict__ norm, int E) {
    int e = blockIdx.x * blockDim.x + threadIdx.x;
    if (e >= E) return;
    float ds = deg[src[e]], dd = deg[dst[e]];
    float is = (ds > 0.0f) ? rsqrtf(fmaxf(ds, 1e-12f)) : 0.0f;
    float id = (dd > 0.0f) ? rsqrtf(fmaxf(dd, 1e-12f)) : 0.0f;
    norm[e] = is * ew[e] * id;
}

// agg[b,t,n,g] = gcn_b[t,g]  (bias pre-seeded so scatter only adds)
__global__ void k_agg_init(float* __restrict__ agg, const float* __restrict__ gcn_b) {
    int idx = blockIdx.x * blockDim.x + threadIdx.x;
    if (idx >= BT * Nn * Gg) return;
    int g = idx & (Gg - 1);
    int t = (idx / (Gg * Nn)) % Tt;
    agg[idx] = gcn_b[t * Gg + g];
}

// fused message: agg[bt,dst,g] += norm[e] * sum_i x[bt,src,i] * gcn_w[t,i,g]
__global__ void k_scatter(const float* __restrict__ x, const float* __restrict__ gcn_w,
                          const int* __restrict__ src, const int* __restrict__ dst,
                          const float* __restrict__ norm, float* __restrict__ agg, int E) {
    long long gid = (long long)blockIdx.x * blockDim.x + threadIdx.x;
    long long total = (long long)E * BT * Gg;
    if (gid >= total) return;
    int g    = (int)(gid & (Gg - 1));
    long long pair = gid >> 5;           // Gg == 32
    int bt   = (int)(pair % BT);
    int e    = (int)(pair / BT);
    int t    = bt % Tt;
    int s    = src[e];
    int d    = dst[e];
    const float* xr = x + ((size_t)bt * Nn + s) * Cc;
    const float* wc = gcn_w + (size_t)t * Cc * Gg + g;
    float acc = 0.0f;
#pragma unroll
    for (int i = 0; i < Cc; ++i) acc += xr[i] * wc[(size_t)i * Gg];
    atomicAdd(&agg[((size_t)bt * Nn + d) * Gg + g], acc * norm[e]);
}

__global__ void k_relu(float* __restrict__ p, int n) {
    int i = blockIdx.x * blockDim.x + threadIdx.x;
    if (i < n) p[i] = fmaxf(p[i], 0.0f);
}

// ---- WMMA tiled GEMM, async double-buffered: out = A(MxK) * W(NxK)^T + b0 + b1 ----
#define BMt 64
#define BNt 64
#define BKt 32
#define LDSK 36   // row stride in floats: rows 16B-aligned (144B), 36*l mod 64 conflict-free

__global__ __launch_bounds__(256)
void k_gemm_bias(const float* __restrict__ A, const float* __restrict__ W,
                 const float* __restrict__ bias0, const float* __restrict__ bias1,
                 float* __restrict__ out, int M, int Nout, int K) {
    __shared__ __align__(16) float As[2][BMt * LDSK];
    __shared__ __align__(16) float Bs[2][BNt * LDSK];

    const int tid  = threadIdx.x;
    const int bm   = blockIdx.y * BMt;
    const int bn   = blockIdx.x * BNt;
    const int wave = tid >> 5;
    const int lane = tid & 31;
    const int lh   = lane & 15;
    const int kh   = (lane >> 4) << 1;          // 0 or 2 (K half-select)
    const int wm   = (wave & 3) * 16;           // 4 wave-rows  -> 64 M
    const int wn   = (wave >> 2) * 32;          // 2 wave-cols, 2 tiles each -> 64 N

    // per-thread copy slots: 2x16B of A-tile, 2x16B of B-tile (fully coalesced)
    const int r0 = tid >> 3;                    // rows 0..31 (and +32)
    const int c0 = (tid & 7) * 4;               // 16B column within 32-float row

    const float* Ab = A + (size_t)bm * K;
    const float* Wb = W + (size_t)bn * K;

    auto issue_tile = [&](int kt, int buf) {
        cp16_to_lds(&As[buf][(size_t)r0 * LDSK + c0],        Ab + (size_t)r0 * K + kt + c0);
        cp16_to_lds(&As[buf][(size_t)(r0 + 32) * LDSK + c0], Ab + (size_t)(r0 + 32) * K + kt + c0);
        cp16_to_lds(&Bs[buf][(size_t)r0 * LDSK + c0],        Wb + (size_t)r0 * K + kt + c0);
        cp16_to_lds(&Bs[buf][(size_t)(r0 + 32) * LDSK + c0], Wb + (size_t)(r0 + 32) * K + kt + c0);
    };

    v8f acc0 = {};
    v8f acc1 = {};

    const int nt = K / BKt;
    issue_tile(0, 0);
    for (int i = 0; i < nt; ++i) {
        const int cur = i & 1;
        if (i + 1 < nt) {
            issue_tile((i + 1) * BKt, cur ^ 1);   // prefetch next tile into other buffer
            wait_async_le<4>();                   // in-order: current tile's 4 copies landed
        } else {
            wait_async_le<0>();
        }
        __syncthreads();
        const float* as = &As[cur][0];
        const float* bs = &Bs[cur][0];
#pragma unroll
        for (int kk = 0; kk < BKt; kk += 4) {
            v2f a  = *(const v2f*)&as[(wm + lh) * LDSK + kk + kh];
            v2f b0 = *(const v2f*)&bs[(wn + lh) * LDSK + kk + kh];
            v2f b1 = *(const v2f*)&bs[(wn + 16 + lh) * LDSK + kk + kh];
            acc0 = wmma_f32(a, b0, acc0);
            acc1 = wmma_f32(a, b1, acc1);
        }
        __syncthreads();
    }

    const int mo = (lane >> 4) * 8;             // C/D layout: lanes 16-31 hold M=8..15
#pragma unroll
    for (int r = 0; r < 8; ++r) {
        int m  = bm + wm + mo + r;
        int n0 = bn + wn + lh;
        int n1 = n0 + 16;
        float bv0 = bias0[n0] + bias1[n0];
        float bv1 = bias0[n1] + bias1[n1];
        out[(size_t)m * Nout + n0] = acc0[r] + bv0;
        out[(size_t)m * Nout + n1] = acc1[r] + bv1;
    }
}

// ---------------- LSTM recurrence step (one wave per 16x16 h-tile) ----------------
// gates = pre[:, t, :] + h_in @ w_hh^T ; cell update done in-lane via C/D layout.
__global__ __launch_bounds__(32)
void k_lstm_step(const float* __restrict__ pre, const float* __restrict__ w_hh,
                 const float* __restrict__ h_in, float* __restrict__ h_out,
                 float* __restrict__ c, float* __restrict__ hs, int t) {
    const int jt   = blockIdx.x * 16;           // hidden tile
    const int mt   = blockIdx.y * 16;           // batch tile
    const int lane = threadIdx.x & 31;
    const int lh   = lane & 15;
    const int kh   = (lane >> 4) << 1;

    v8f ai = {}, af = {}, ag = {}, ao = {};
    const float* wi = w_hh + (size_t)(0 * Hh + jt + lh) * Hh;
    const float* wf = w_hh + (size_t)(1 * Hh + jt + lh) * Hh;
    const float* wg = w_hh + (size_t)(2 * Hh + jt + lh) * Hh;
    const float* wo = w_hh + (size_t)(3 * Hh + jt + lh) * Hh;
    const float* ha = h_in + (size_t)(mt + lh) * Hh;

    for (int k = 0; k < Hh; k += 4) {
        v2f a  = *(const v2f*)(ha + k + kh);
        v2f bi = *(const v2f*)(wi + k + kh);
        v2f bf = *(const v2f*)(wf + k + kh);
        v2f bg = *(const v2f*)(wg + k + kh);
        v2f bo = *(const v2f*)(wo + k + kh);
        ai = wmma_f32(a, bi, ai);
        af = wmma_f32(a, bf, af);
        ag = wmma_f32(a, bg, ag);
        ao = wmma_f32(a, bo, ao);
    }

    const int mo = (lane >> 4) * 8;
    const int n  = jt + lh;
#pragma unroll
    for (int r = 0; r < 8; ++r) {
        int b = mt + mo + r;                            // batch index
        size_t prow = ((size_t)b * Tt + t) * H4;        // pre row (b,t)
        float gi = ai[r] + pre[prow + 0 * Hh + n];
        float gf = af[r] + pre[prow + 1 * Hh + n];
        float gg = ag[r] + pre[prow + 2 * Hh + n];
        float go = ao[r] + pre[prow + 3 * Hh + n];
        float is = sigm(gi), fs = sigm(gf), os = sigm(go);
        float gt = tanhf(gg);
        size_t hc = (size_t)b * Hh + n;
        float cn = fs * c[hc] + is * gt;
        float hn = os * tanhf(cn);
        c[hc] = cn;
        h_out[hc] = hn;
        if (hs) hs[((size_t)b * Tt + t) * Hh + n] = hn;
    }
}

// ---------------- final projection: (32x512) @ (8x512)^T + b ----------------
__global__ void k_proj(const float* __restrict__ h, const float* __restrict__ pw,
                       const float* __restrict__ pb, float* __restrict__ out) {
    int tid = threadIdx.x;                       // 256 threads, one per (b,o)
    int b = tid >> 3, o = tid & 7;
    const float* hr = h + (size_t)b * Hh;
    const float* wr = pw + (size_t)o * Hh;
    float acc = 0.0f;
    for (int k = 0; k < Hh; ++k) acc += hr[k] * wr[k];
    out[b * Oo + o] = acc + pb[o];
}

// ---------------- host orchestration ----------------
extern "C" void kernel_launch(void* const* d_in, const int* in_sizes, int n_in,
                              void* d_out, int out_size, void* d_ws, size_t ws_size,
                              hipStream_t stream) {
    const float* x      = (const float*)d_in[0];
    const int*   ei     = (const int*)d_in[1];
    const float* ew     = (const float*)d_in[2];
    const float* gcn_w  = (const float*)d_in[3];
    const float* gcn_b  = (const float*)d_in[4];
    const float* w_ih0  = (const float*)d_in[5];
    const float* w_hh0  = (const float*)d_in[6];
    const float* b_ih0  = (const float*)d_in[7];
    const float* b_hh0  = (const float*)d_in[8];
    const float* w_ih1  = (const float*)d_in[9];
    const float* w_hh1  = (const float*)d_in[10];
    const float* b_ih1  = (const float*)d_in[11];
    const float* b_hh1  = (const float*)d_in[12];
    const float* proj_w = (const float*)d_in[13];
    const float* proj_b = (const float*)d_in[14];
    float*       out    = (float*)d_out;

    const int E = in_sizes[2];                   // 9000
    const int* e_src = ei;
    const int* e_dst = ei + E;

    // workspace carve-up (floats)
    float* ws   = (float*)d_ws;
    float* seq  = ws;                                          // 16,384,000
    float* pre0 = seq  + (size_t)BT * NG;                      //  1,048,576
    float* out1 = pre0 + (size_t)BT * H4;                      //    262,144
    float* pre1 = out1 + (size_t)BT * Hh;                      //  1,048,576
    float* hc   = pre1 + (size_t)BT * H4;                      //  6 * 16384
    float* h0A  = hc;
    float* h0B  = hc + 1 * Bb * Hh;
    float* c0   = hc + 2 * Bb * Hh;
    float* h1A  = hc + 3 * Bb * Hh;
    float* h1B  = hc + 4 * Bb * Hh;
    float* c1   = hc + 5 * Bb * Hh;
    float* deg  = hc + 6 * Bb * Hh;                            //       1024
    float* nrm  = deg + 1024;                                  //       E

    const int THR = 256;

    // --- GCN normalization ---
    k_zero<<<(1024 + THR - 1) / THR, THR, 0, stream>>>(deg, 1024);
    k_deg<<<(E + THR - 1) / THR, THR, 0, stream>>>(e_dst, ew, deg, E);
    k_norm<<<(E + THR - 1) / THR, THR, 0, stream>>>(e_src, e_dst, ew, deg, nrm, E);

    // --- GCN aggregate: seq = relu(scatter(x @ gcn_w) + bias) ---
    int nAgg = BT * Nn * Gg;
    k_agg_init<<<(nAgg + THR - 1) / THR, THR, 0, stream>>>(seq, gcn_b);
    long long nMsg = (long long)E * BT * Gg;
    k_scatter<<<(unsigned)((nMsg + THR - 1) / THR), THR, 0, stream>>>(
        x, gcn_w, e_src, e_dst, nrm, seq, E);
    k_relu<<<(nAgg + THR - 1) / THR, THR, 0, stream>>>(seq, nAgg);

    // --- zero h/c state ---
    k_zero<<<(6 * Bb * Hh + THR - 1) / THR, THR, 0, stream>>>(hc, 6 * Bb * Hh);

    // --- LSTM layer 0 ---
    dim3 g0(H4 / BNt, BT / BMt);                     // (32, 8)
    k_gemm_bias<<<g0, 256, 0, stream>>>(seq, w_ih0, b_ih0, b_hh0, pre0, BT, H4, NG);
    {
        float* cur = h0A; float* nxt = h0B;
        dim3 gs(Hh / 16, Bb / 16);                   // (32, 2)
        for (int t = 0; t < Tt; ++t) {
            k_lstm_step<<<gs, 32, 0, stream>>>(pre0, w_hh0, cur, nxt, c0, out1, t);
            float* tmp = cur; cur = nxt; nxt = tmp;
        }
    }

    // --- LSTM layer 1 ---
    k_gemm_bias<<<g0, 256, 0, stream>>>(out1, w_ih1, b_ih1, b_hh1, pre1, BT, H4, Hh);
    float* hFinal;
    {
        float* cur = h1A; float* nxt = h1B;
        dim3 gs(Hh / 16, Bb / 16);
        for (int t = 0; t < Tt; ++t) {
            k_lstm_step<<<gs, 32, 0, stream>>>(pre1, w_hh1, cur, nxt, c1, nullptr, t);
            float* tmp = cur; cur = nxt; nxt = tmp;
        }
        hFinal = cur;                                // after even #swaps = h1A
    }

    // --- projection ---
    k_proj<<<1, 256, 0, stream>>>(hFinal, proj_w, proj_b, out);
}